// SGC_66709432041921
// MI455X (gfx1250) — compile-verified
//
#include <hip/hip_runtime.h>

#define N_NODES 100000
#define N_EDGES 1200000
#define IN_FEATS 64
#define OUT_FEATS 32
#define K_HOPS 4

typedef __attribute__((ext_vector_type(2))) float v2f;
typedef __attribute__((ext_vector_type(8))) float v8f;

// ---------------- degree / norm / edge-weight precompute ----------------

__global__ void sgc_degree_kernel(const int* __restrict__ dst,
                                  float* __restrict__ deg, int n_edges) {
    int e = blockIdx.x * blockDim.x + threadIdx.x;
    if (e < n_edges) atomicAdd(&deg[dst[e]], 1.0f);
}

__global__ void sgc_norm_kernel(float* __restrict__ deg_norm, int n_nodes) {
    int i = blockIdx.x * blockDim.x + threadIdx.x;
    if (i < n_nodes) {
        float d = deg_norm[i];
        deg_norm[i] = rsqrtf(fmaxf(d, 1.0f));
    }
}

__global__ void sgc_edgew_kernel(const float* __restrict__ norm,
                                 const int* __restrict__ src,
                                 const int* __restrict__ dst,
                                 float* __restrict__ ew, int n_edges) {
    int e = blockIdx.x * blockDim.x + threadIdx.x;
    if (e < n_edges) ew[e] = norm[src[e]] * norm[dst[e]];
}

// ---------------- weighted scatter-add (one wave per edge) ----------------
// acc[dst[e]][f] += h[src[e]][f] * w[e]   for f = 0..63, 2 floats per lane.

__global__ void sgc_scatter_kernel(const float* __restrict__ h,
                                   const float* __restrict__ ew,
                                   const int* __restrict__ src,
                                   const int* __restrict__ dst,
                                   float* __restrict__ acc, int n_edges) {
    int t = blockIdx.x * blockDim.x + threadIdx.x;
    int e = t >> 5;            // wave-uniform edge id
    if (e >= n_edges) return;
    int lane = t & 31;
    int s = src[e];
    int d = dst[e];
    float w = ew[e];
    const float* hp = h + (size_t)s * IN_FEATS + lane * 2;
    float* ap = acc + (size_t)d * IN_FEATS + lane * 2;
    float2 v = *(const float2*)hp;
    atomicAdd(ap + 0, v.x * w);
    atomicAdd(ap + 1, v.y * w);
}

// ---------------- final linear layer via f32 WMMA ----------------
// out[m][n] = sum_k h[m][k] * W[n][k] + b[n]
// One wave computes a 16-row tile; N=32 -> two 16x16x4 WMMA chains of K=64.

__global__ void sgc_gemm_wmma_kernel(const float* __restrict__ h,
                                     const float* __restrict__ W,
                                     const float* __restrict__ b,
                                     float* __restrict__ out, int n_tiles) {
    int wave = (int)((blockIdx.x * blockDim.x + threadIdx.x) >> 5);
    if (wave >= n_tiles) return;           // wave-uniform: EXEC stays all-1s
    int lane = threadIdx.x & 31;
    int l15  = lane & 15;
    int hi   = lane >> 4;                  // 0: K=0,1  1: K=2,3 (A layout)

    // A: lane<16 holds row M=l15 {K=k,k+1}; lane>=16 holds row M=l15 {K=k+2,k+3}
    const float* arow = h + (size_t)(wave * 16 + l15) * IN_FEATS + hi * 2;

    #pragma unroll
    for (int t = 0; t < 2; ++t) {
        int n = t * 16 + l15;
        // B = W^T: B[k][n] = W[n][k]; lane layout mirrors A (contiguous in W row)
        const float* brow = W + (size_t)n * IN_FEATS + hi * 2;
        float bias = b[n];
        v8f c = {bias, bias, bias, bias, bias, bias, bias, bias};

        #pragma unroll
        for (int k = 0; k < IN_FEATS; k += 4) {
            v2f a  = *(const v2f*)(arow + k);
            v2f bb = *(const v2f*)(brow + k);
            c = __builtin_amdgcn_wmma_f32_16x16x4_f32(
                    /*neg_a=*/false, a, /*neg_b=*/false, bb,
                    /*c_mod=*/(short)0, c, /*reuse_a=*/false, /*reuse_b=*/false);
        }

        // D layout: VGPR r -> (M = r + hi*8, N = l15) per 16x16 f32 C/D layout
        int m0 = wave * 16 + hi * 8;
        #pragma unroll
        for (int r = 0; r < 8; ++r) {
            out[(size_t)(m0 + r) * OUT_FEATS + n] = c[r];
        }
    }
}

// ---------------- launch ----------------

static inline size_t align256(size_t x) { return (x + 255) & ~(size_t)255; }

extern "C" void kernel_launch(void* const* d_in, const int* in_sizes, int n_in,
                              void* d_out, int out_size, void* d_ws, size_t ws_size,
                              hipStream_t stream) {
    const float* features = (const float*)d_in[0];
    const int*   src      = (const int*)d_in[1];
    const int*   dst      = (const int*)d_in[2];
    const float* W        = (const float*)d_in[3];
    const float* b        = (const float*)d_in[4];
    float*       out      = (float*)d_out;

    char* ws = (char*)d_ws;
    size_t off = 0;
    float* norm = (float*)(ws + off); off += align256((size_t)N_NODES * 4);
    float* ew   = (float*)(ws + off); off += align256((size_t)N_EDGES * 4);
    float* bufA = (float*)(ws + off); off += align256((size_t)N_NODES * IN_FEATS * 4);
    float* bufB = (float*)(ws + off); off += align256((size_t)N_NODES * IN_FEATS * 4);
    (void)ws_size; (void)in_sizes; (void)n_in; (void)out_size;

    // degree -> norm -> per-edge weight
    hipMemsetAsync(norm, 0, (size_t)N_NODES * 4, stream);
    sgc_degree_kernel<<<(N_EDGES + 255) / 256, 256, 0, stream>>>(dst, norm, N_EDGES);
    sgc_norm_kernel<<<(N_NODES + 255) / 256, 256, 0, stream>>>(norm, N_NODES);
    sgc_edgew_kernel<<<(N_EDGES + 255) / 256, 256, 0, stream>>>(norm, src, dst, ew, N_EDGES);

    // K_HOPS weighted scatter-adds, ping-pong bufA/bufB
    const float* hprev = features;
    float* bufs[2] = {bufA, bufB};
    long long scatter_threads = (long long)N_EDGES * 32;
    int scatter_blocks = (int)((scatter_threads + 255) / 256);
    for (int hop = 0; hop < K_HOPS; ++hop) {
        float* acc = bufs[hop & 1];
        hipMemsetAsync(acc, 0, (size_t)N_NODES * IN_FEATS * 4, stream);
        sgc_scatter_kernel<<<scatter_blocks, 256, 0, stream>>>(
            hprev, ew, src, dst, acc, N_EDGES);
        hprev = acc;
    }

    // final linear layer with f32 WMMA (one wave per 16-node tile)
    int n_tiles = N_NODES / 16;                       // 100000 % 16 == 0
    int waves_per_block = 8;                          // 256 threads
    int gemm_blocks = (n_tiles + waves_per_block - 1) / waves_per_block;
    sgc_gemm_wmma_kernel<<<gemm_blocks, 256, 0, stream>>>(hprev, W, b, out, n_tiles);
}